// DIFFormerConv_68685116997931
// MI455X (gfx1250) — compile-verified
//
#include <hip/hip_runtime.h>
#include <hip/hip_bf16.h>

typedef __attribute__((ext_vector_type(2))) float v2f;
typedef __attribute__((ext_vector_type(8))) float v8f;

// workspace layout (floats):
//   g_qn  : n*256
//   g_red : kvs[4*64*64]=16384, vs_sum[256], ks_sum[256]  (16896 total)

__global__ void diff_zero_red(float* __restrict__ red) {
    int g = blockIdx.x * 256 + threadIdx.x;
    if (g < 16896) red[g] = 0.f;
}

// ---------------- kernel 1: QKV GEMM + normalize + reductions ----------------
// grid: ceil(n/64) blocks of 256 threads (8 waves). 256KB dynamic LDS.
__global__ void diff_qkv(const float* __restrict__ x,
                         const float* __restrict__ Wq, const float* __restrict__ bq,
                         const float* __restrict__ Wk, const float* __restrict__ bk,
                         const float* __restrict__ Wv, const float* __restrict__ bv,
                         float* __restrict__ g_qn, float* __restrict__ g_red, int n) {
    extern __shared__ float sm[];
    float* sm_x = sm;               // 64*256
    float* sm_q = sm + 16384;
    float* sm_k = sm + 2 * 16384;
    float* sm_v = sm + 3 * 16384;

    const int tid  = threadIdx.x;
    const int wave = tid >> 5, lane = tid & 31;
    const int half = lane >> 4, nn = lane & 15;
    const long row0 = (long)blockIdx.x * 64;

    // stage x tile [64 x 256] into LDS (zero-pad ragged tail)
    for (int i = 0; i < 16; ++i) {
        int e4 = tid + i * 256;           // float4 index, 4096 total
        int r = e4 >> 6, c4 = e4 & 63;
        float4 val = make_float4(0.f, 0.f, 0.f, 0.f);
        if (row0 + r < n) val = ((const float4*)(x + (row0 + r) * 256))[c4];
        ((float4*)sm_x)[e4] = val;
    }
    __syncthreads();

    // mat loop fully unrolled with compile-time pointer selection so the
    // weight/bias loads stay in the GLOBAL address space (LOADcnt only),
    // not FLAT (which would also tick DScnt and serialize against LDS).
#pragma unroll
    for (int mat = 0; mat < 3; ++mat) {
        const float* W    = (mat == 0) ? Wq : (mat == 1) ? Wk : Wv;
        const float* Bias = (mat == 0) ? bq : (mat == 1) ? bk : bv;
        float* S          = (mat == 0) ? sm_q : (mat == 1) ? sm_k : sm_v;

        for (int ct = wave; ct < 16; ct += 8) {   // 2 col-tiles per wave per mat
            v8f acc[4];
            acc[0] = {}; acc[1] = {}; acc[2] = {}; acc[3] = {};
            const float* wrow = W + (ct * 16 + nn) * 256 + 2 * half;  // B[k][o]=W[o][k]
            const float* ab   = sm_x + nn * 256 + 2 * half;           // A[m][k]=x row
            for (int k0 = 0; k0 < 256; k0 += 4) {
                v2f b = *(const v2f*)(wrow + k0);
#pragma unroll
                for (int rt = 0; rt < 4; ++rt) {
                    v2f a = *(const v2f*)(ab + rt * 4096 + k0);
                    acc[rt] = __builtin_amdgcn_wmma_f32_16x16x4_f32(
                        false, a, false, b, (short)0, acc[rt], false, false);
                }
            }
            const float bias = Bias[ct * 16 + nn];
#pragma unroll
            for (int rt = 0; rt < 4; ++rt)
#pragma unroll
                for (int j = 0; j < 8; ++j)
                    S[(rt * 16 + j + 8 * half) * 256 + ct * 16 + nn] = acc[rt][j] + bias;
        }
    }
    __syncthreads();

    // per-(row,head) L2 normalize q,k; zero padded rows (incl. v) so reductions stay clean
    {
        const int r = tid >> 2, h = tid & 3;
        const bool live = (row0 + r) < n;
        float* qp = sm_q + r * 256 + h * 64;
        float* kp = sm_k + r * 256 + h * 64;
        float* vp = sm_v + r * 256 + h * 64;
        float sq = 0.f, sk = 0.f;
        for (int d = 0; d < 64; ++d) { sq += qp[d] * qp[d]; sk += kp[d] * kp[d]; }
        const float rq = live ? rsqrtf(sq) : 0.f;
        const float rk = live ? rsqrtf(sk) : 0.f;
        for (int d = 0; d < 64; ++d) {
            qp[d] *= rq; kp[d] *= rk;
            if (!live) vp[d] = 0.f;
        }
    }
    __syncthreads();

    // write qn tile to workspace
    for (int i = 0; i < 16; ++i) {
        int e4 = tid + i * 256;
        int r = e4 >> 6, c4 = e4 & 63;
        if (row0 + r < n)
            ((float4*)(g_qn + (row0 + r) * 256))[c4] = ((const float4*)sm_q)[e4];
    }

    // ks_sum / vs_sum partials (one (h,d) column per thread)
    {
        float sk = 0.f, sv = 0.f;
        for (int r = 0; r < 64; ++r) {
            sk += sm_k[r * 256 + tid];
            sv += sm_v[r * 256 + tid];
        }
        atomicAdd(g_red + 16384 + tid, sv);        // vs_sum
        atomicAdd(g_red + 16640 + tid, sk);        // ks_sum
    }

    // kvs partial: thread t -> (h, m); 64-wide outer product rows accumulated locally
    {
        const int h = tid >> 6, m = tid & 63;
        float4 acc[16];
#pragma unroll
        for (int d4 = 0; d4 < 16; ++d4) acc[d4] = make_float4(0.f, 0.f, 0.f, 0.f);
        for (int r = 0; r < 64; ++r) {
            const float kvv = sm_k[r * 256 + h * 64 + m];
            const float4* vp = (const float4*)(sm_v + r * 256 + h * 64);
#pragma unroll
            for (int d4 = 0; d4 < 16; ++d4) {
                float4 vv = vp[d4];
                acc[d4].x += kvv * vv.x; acc[d4].y += kvv * vv.y;
                acc[d4].z += kvv * vv.z; acc[d4].w += kvv * vv.w;
            }
        }
        float* kb = g_red + h * 4096 + m * 64;
#pragma unroll
        for (int d4 = 0; d4 < 16; ++d4) {
            atomicAdd(kb + d4 * 4 + 0, acc[d4].x);
            atomicAdd(kb + d4 * 4 + 1, acc[d4].y);
            atomicAdd(kb + d4 * 4 + 2, acc[d4].z);
            atomicAdd(kb + d4 * 4 + 3, acc[d4].w);
        }
    }
}

// ---------------- kernel 2: num = qn @ kvs, denom, head mean ----------------
// grid: ceil(n/128) blocks of 256 threads (8 waves, 16 rows each). ~196KB LDS.
__global__ void diff_out(const float* __restrict__ g_qn,
                         const float* __restrict__ g_red,
                         float* __restrict__ out, int n) {
    extern __shared__ float sm[];
    float* sm_kvs = sm;                    // 16384
    float* sm_vs  = sm + 16384;            // 256
    float* sm_ks  = sm + 16640;            // 256
    float* sm_qn  = sm + 16896;            // 128*256
    float* sm_den = sm + 16896 + 32768;    // 128*4

    const int tid  = threadIdx.x;
    const int wave = tid >> 5, lane = tid & 31;
    const int half = lane >> 4, nn = lane & 15;
    const long row0 = (long)blockIdx.x * 128;

    for (int i = 0; i < 16; ++i)
        ((float4*)sm_kvs)[tid + i * 256] = ((const float4*)g_red)[tid + i * 256];
    sm_vs[tid] = g_red[16384 + tid];
    sm_ks[tid] = g_red[16640 + tid];
    for (int i = 0; i < 32; ++i) {
        int e4 = tid + i * 256;            // 8192 float4
        int r = e4 >> 6, c4 = e4 & 63;
        float4 v = make_float4(0.f, 0.f, 0.f, 0.f);
        if (row0 + r < n) v = ((const float4*)(g_qn + (row0 + r) * 256))[c4];
        ((float4*)sm_qn)[e4] = v;
    }
    __syncthreads();

    const float fn = (float)n;
    for (int p = tid; p < 512; p += 256) {          // denom per (row, head)
        int r = p >> 2, h = p & 3;
        const float* qp = sm_qn + r * 256 + h * 64;
        const float* kp = sm_ks + h * 64;
        float s = 0.f;
        for (int d = 0; d < 64; ++d) s += qp[d] * kp[d];
        sm_den[p] = s + fn;
    }
    __syncthreads();

    const int lr0 = wave * 16;
#pragma unroll
    for (int c = 0; c < 4; ++c) {
        v8f oacc = {};
#pragma unroll
        for (int h = 0; h < 4; ++h) {
            v8f acc = {};
            const float* qb = sm_qn + (lr0 + nn) * 256 + h * 64 + 2 * half;
            const float* kb = sm_kvs + h * 4096 + 2 * half * 64 + c * 16 + nn;
#pragma unroll
            for (int k0 = 0; k0 < 64; k0 += 4) {
                v2f a = *(const v2f*)(qb + k0);
                v2f b;
                b.x = kb[k0 * 64];
                b.y = kb[k0 * 64 + 64];
                acc = __builtin_amdgcn_wmma_f32_16x16x4_f32(
                    false, a, false, b, (short)0, acc, false, false);
            }
            const int d = c * 16 + nn;
            const float vs = sm_vs[h * 64 + d];
#pragma unroll
            for (int j = 0; j < 8; ++j) {
                int lr = lr0 + j + 8 * half;
                oacc[j] += (acc[j] + vs) / sm_den[lr * 4 + h];
            }
        }
#pragma unroll
        for (int j = 0; j < 8; ++j) {
            long gr = row0 + lr0 + j + 8 * half;
            if (gr < n) out[gr * 64 + c * 16 + nn] = oacc[j] * 0.25f;
        }
    }
}

extern "C" void kernel_launch(void* const* d_in, const int* in_sizes, int n_in,
                              void* d_out, int out_size, void* d_ws, size_t ws_size,
                              hipStream_t stream) {
    const float* x   = (const float*)d_in[0];
    const float* Wq  = (const float*)d_in[1];
    const float* bq  = (const float*)d_in[2];
    const float* Wk  = (const float*)d_in[3];
    const float* bk  = (const float*)d_in[4];
    const float* Wv  = (const float*)d_in[5];
    const float* bv  = (const float*)d_in[6];
    float* out = (float*)d_out;
    const int n = in_sizes[0] / 256;

    float* ws_f  = (float*)d_ws;
    float* g_qn  = ws_f;                       // n*256 floats
    float* g_red = ws_f + (size_t)n * 256;     // 16896 floats

    const int qkv_lds = 4 * 16384 * 4;              // 256 KB
    const int out_lds = (16896 + 32768 + 512) * 4;  // ~196 KB
    hipFuncSetAttribute((const void*)diff_qkv, hipFuncAttributeMaxDynamicSharedMemorySize, qkv_lds);
    hipFuncSetAttribute((const void*)diff_out, hipFuncAttributeMaxDynamicSharedMemorySize, out_lds);

    diff_zero_red<<<(16896 + 255) / 256, 256, 0, stream>>>(g_red);
    diff_qkv<<<(n + 63) / 64, 256, qkv_lds, stream>>>(x, Wq, bq, Wk, bk, Wv, bv,
                                                      g_qn, g_red, n);
    diff_out<<<(n + 127) / 128, 256, out_lds, stream>>>(g_qn, g_red, out, n);
}